// BondConvSum_80917183856995
// MI455X (gfx1250) — compile-verified
//
#include <hip/hip_runtime.h>
#include <hip/hip_bf16.h>
#include <math.h>

// ---------------------------------------------------------------------------
// BondConvSum fused pipeline for MI455X (gfx1250), wave32, WMMA f32 16x16x4.
// ---------------------------------------------------------------------------

#define ATOM  128
#define BOND  128
#define ANGLE 64
#define BN_EPS 1e-5f

typedef __attribute__((ext_vector_type(2))) float v2f;
typedef __attribute__((ext_vector_type(8))) float v8f;

// D = A(16x4) * B(4x16) + C, fp32 WMMA. 8-arg form:
// (neg_a, A, neg_b, B, c_mod, C, reuse_a, reuse_b)
__device__ __forceinline__ v8f wmma4(v2f a, v2f b, v8f c) {
    return __builtin_amdgcn_wmma_f32_16x16x4_f32(
        false, a, false, b, (short)0, c, false, false);
}

__device__ __forceinline__ v2f ld2(const float* p) {        // 8B-aligned b64 load
    return *(const v2f*)p;
}

// ---------------------------------------------------------------------------
// Kernel 1: vertex projections.  VP[n][0:128]=v@ws_c^T, [128:256]=v@wd_c^T,
//                                [256:384]=v@ws_g^T,   [384:512]=v@wd_g^T
// One block = 16 rows of vertex_feat; wave w computes n-tile w (8 waves).
// Weight loop fully unrolled: 1 A-frag feeds 4 WMMAs, all-global B loads.
// ---------------------------------------------------------------------------
__global__ void node_proj_kernel(const float* __restrict__ vf,
                                 const float* __restrict__ w0,
                                 const float* __restrict__ w1,
                                 const float* __restrict__ w2,
                                 const float* __restrict__ w3,
                                 float* __restrict__ VP, int N) {
    __shared__ float As[16 * 130];              // padded: stride 130 avoids bank conflicts
    const int m0   = blockIdx.x * 16;
    const int tid  = threadIdx.x;               // 0..255
    const int wave = tid >> 5, lane = tid & 31;

    for (int i = tid; i < 16 * ATOM; i += 256) {
        int r = i >> 7, c = i & 127;
        int row = m0 + r; if (row >= N) row = N - 1;
        As[r * 130 + c] = vf[(size_t)row * ATOM + c];
    }
    __syncthreads();

    const int n0 = wave * 16;
    const int mlo = lane & 15, khalf = lane >> 4;
    const size_t wrow = (size_t)(n0 + mlo) * ATOM;   // B[k][n] = W[n][k]

    v8f a0 = {}, a1 = {}, a2 = {}, a3 = {};
    #pragma unroll
    for (int k = 0; k < ATOM; k += 4) {
        const int kk = k + 2 * khalf;
        v2f a  = ld2(&As[mlo * 130 + kk]);
        v2f b0 = ld2(w0 + wrow + kk);
        v2f b1 = ld2(w1 + wrow + kk);
        v2f b2 = ld2(w2 + wrow + kk);
        v2f b3 = ld2(w3 + wrow + kk);
        a0 = wmma4(a, b0, a0);
        a1 = wmma4(a, b1, a1);
        a2 = wmma4(a, b2, a2);
        a3 = wmma4(a, b3, a3);
    }
    #pragma unroll
    for (int v = 0; v < 8; ++v) {
        const int m = m0 + v + 8 * khalf;
        if (m < N) {
            float* dst = VP + (size_t)m * 512 + n0 + mlo;
            dst[0]   = a0[v];
            dst[128] = a1[v];
            dst[256] = a2[v];
            dst[384] = a3[v];
        }
    }
}

// ---------------------------------------------------------------------------
// Kernel 2: edge projections.  EP[e][0:128]=e@wb_c^T, [128:256]=e@wb_g^T
// ---------------------------------------------------------------------------
__global__ void edge_proj_kernel(const float* __restrict__ ef,
                                 const float* __restrict__ w0,
                                 const float* __restrict__ w1,
                                 float* __restrict__ EP, int E) {
    __shared__ float As[16 * 130];
    const int m0   = blockIdx.x * 16;
    const int tid  = threadIdx.x;
    const int wave = tid >> 5, lane = tid & 31;

    for (int i = tid; i < 16 * BOND; i += 256) {
        int r = i >> 7, c = i & 127;
        int row = m0 + r; if (row >= E) row = E - 1;
        As[r * 130 + c] = ef[(size_t)row * BOND + c];
    }
    __syncthreads();

    const int n0 = wave * 16;
    const int mlo = lane & 15, khalf = lane >> 4;
    const size_t wrow = (size_t)(n0 + mlo) * BOND;

    v8f a0 = {}, a1 = {};
    #pragma unroll
    for (int k = 0; k < BOND; k += 4) {
        const int kk = k + 2 * khalf;
        v2f a  = ld2(&As[mlo * 130 + kk]);
        v2f b0 = ld2(w0 + wrow + kk);
        v2f b1 = ld2(w1 + wrow + kk);
        a0 = wmma4(a, b0, a0);
        a1 = wmma4(a, b1, a1);
    }
    #pragma unroll
    for (int v = 0; v < 8; ++v) {
        const int m = m0 + v + 8 * khalf;
        if (m < E) {
            float* dst = EP + (size_t)m * 256 + n0 + mlo;
            dst[0]   = a0[v];
            dst[128] = a1[v];
        }
    }
}

// ---------------------------------------------------------------------------
// Kernel 3: pass 1 over T — angle GEMM + gathers, accumulate BN statistics.
// stats[0:128]=sum_core [128:256]=sumsq_core [256:384]=sum_gate [384:512]=sumsq_gate
// ---------------------------------------------------------------------------
__global__ void pass1_kernel(const float* __restrict__ af,
                             const int* __restrict__ kidx,
                             const int* __restrict__ jidx,
                             const int* __restrict__ iidx,
                             const float* __restrict__ wac,
                             const float* __restrict__ wag,
                             const float* __restrict__ VP,
                             const float* __restrict__ EP,
                             float* __restrict__ stats, int T) {
    __shared__ float As[16 * 66];
    __shared__ int Lk[16], Lj[16], Li[16];
    __shared__ float lsum[512];

    const int tid  = threadIdx.x;
    const int wave = tid >> 5, lane = tid & 31;
    const int n0   = wave * 16;
    const int mlo  = lane & 15, khalf = lane >> 4;
    const int n    = n0 + mlo;
    const size_t wrow = (size_t)n * ANGLE;

    for (int i = tid; i < 512; i += 256) lsum[i] = 0.0f;
    __syncthreads();

    const int nTiles = (T + 15) >> 4;
    for (int tile = blockIdx.x; tile < nTiles; tile += gridDim.x) {
        const int t0 = tile * 16;
        for (int i = tid; i < 16 * ANGLE; i += 256) {
            int r = i >> 6, c = i & 63;
            As[r * 66 + c] = (t0 + r < T) ? af[(size_t)(t0 + r) * ANGLE + c] : 0.0f;
        }
        if (tid < 16) {
            int t = t0 + tid; if (t >= T) t = T - 1;
            Lk[tid] = kidx[t]; Lj[tid] = jidx[t]; Li[tid] = iidx[t];
        }
        __syncthreads();

        v8f cc = {}, gg = {};
        #pragma unroll
        for (int k = 0; k < ANGLE; k += 4) {
            const int kk = k + 2 * khalf;
            v2f a  = ld2(&As[mlo * 66 + kk]);
            v2f bc = ld2(wac + wrow + kk);
            v2f bg = ld2(wag + wrow + kk);
            cc = wmma4(a, bc, cc);
            gg = wmma4(a, bg, gg);
        }

        float sc = 0.f, qc = 0.f, sg = 0.f, qg = 0.f;
        #pragma unroll
        for (int v = 0; v < 8; ++v) {
            const int m = v + 8 * khalf;
            if (t0 + m < T) {
                const size_t jj = (size_t)Lj[m] * 512;
                const size_t ii = (size_t)Li[m] * 512;
                const size_t ke = (size_t)Lk[m] * 256;
                float c1 = cc[v] + VP[jj + n] + VP[ii + 128 + n] + EP[ke + n];
                float g1 = gg[v] + VP[jj + 256 + n] + VP[ii + 384 + n] + EP[ke + 128 + n];
                sc += c1; qc += c1 * c1;
                sg += g1; qg += g1 * g1;
            }
        }
        atomicAdd(&lsum[n],       sc);
        atomicAdd(&lsum[128 + n], qc);
        atomicAdd(&lsum[256 + n], sg);
        atomicAdd(&lsum[384 + n], qg);
        __syncthreads();
    }
    for (int i = tid; i < 512; i += 256) atomicAdd(&stats[i], lsum[i]);
}

// ---------------------------------------------------------------------------
// Kernel 4: fold BN stats into affine scale/shift.
// bnp[0:128]=scale_c [128:256]=shift_c [256:384]=scale_g [384:512]=shift_g
// ---------------------------------------------------------------------------
__global__ void finalize_kernel(const float* __restrict__ stats,
                                const float* __restrict__ gam_c,
                                const float* __restrict__ bet_c,
                                const float* __restrict__ gam_g,
                                const float* __restrict__ bet_g,
                                float* __restrict__ bnp, float invT) {
    const int nidx = threadIdx.x;   // 128 threads
    float mc = stats[nidx] * invT;
    float vc = stats[128 + nidx] * invT - mc * mc;
    float sc = gam_c[nidx] * rsqrtf(vc + BN_EPS);
    bnp[nidx]       = sc;
    bnp[128 + nidx] = bet_c[nidx] - mc * sc;

    float mg = stats[256 + nidx] * invT;
    float vg = stats[384 + nidx] * invT - mg * mg;
    float sg = gam_g[nidx] * rsqrtf(vg + BN_EPS);
    bnp[256 + nidx] = sg;
    bnp[384 + nidx] = bet_g[nidx] - mg * sg;
}

// ---------------------------------------------------------------------------
// Kernel 5: pass 2 — recompute, BN + silu/sigmoid gate, segment-sum scatter.
// ---------------------------------------------------------------------------
__global__ void pass2_kernel(const float* __restrict__ af,
                             const int* __restrict__ kidx,
                             const int* __restrict__ jidx,
                             const int* __restrict__ iidx,
                             const float* __restrict__ wac,
                             const float* __restrict__ wag,
                             const float* __restrict__ VP,
                             const float* __restrict__ EP,
                             const float* __restrict__ bnp,
                             float* __restrict__ S, int T) {
    __shared__ float As[16 * 66];
    __shared__ int Lk[16], Lj[16], Li[16];
    __shared__ float bn[512];

    const int tid  = threadIdx.x;
    const int wave = tid >> 5, lane = tid & 31;
    const int n0   = wave * 16;
    const int mlo  = lane & 15, khalf = lane >> 4;
    const int n    = n0 + mlo;
    const size_t wrow = (size_t)n * ANGLE;

    for (int i = tid; i < 512; i += 256) bn[i] = bnp[i];
    __syncthreads();
    const float scale_c = bn[n],       shift_c = bn[128 + n];
    const float scale_g = bn[256 + n], shift_g = bn[384 + n];

    const int nTiles = (T + 15) >> 4;
    for (int tile = blockIdx.x; tile < nTiles; tile += gridDim.x) {
        const int t0 = tile * 16;
        for (int i = tid; i < 16 * ANGLE; i += 256) {
            int r = i >> 6, c = i & 63;
            As[r * 66 + c] = (t0 + r < T) ? af[(size_t)(t0 + r) * ANGLE + c] : 0.0f;
        }
        if (tid < 16) {
            int t = t0 + tid; if (t >= T) t = T - 1;
            Lk[tid] = kidx[t]; Lj[tid] = jidx[t]; Li[tid] = iidx[t];
        }
        __syncthreads();

        v8f cc = {}, gg = {};
        #pragma unroll
        for (int k = 0; k < ANGLE; k += 4) {
            const int kk = k + 2 * khalf;
            v2f a  = ld2(&As[mlo * 66 + kk]);
            v2f bc = ld2(wac + wrow + kk);
            v2f bg = ld2(wag + wrow + kk);
            cc = wmma4(a, bc, cc);
            gg = wmma4(a, bg, gg);
        }

        #pragma unroll
        for (int v = 0; v < 8; ++v) {
            const int m = v + 8 * khalf;
            if (t0 + m < T) {
                const size_t jj = (size_t)Lj[m] * 512;
                const size_t ii = (size_t)Li[m] * 512;
                const size_t ke = (size_t)Lk[m] * 256;
                float c1 = cc[v] + VP[jj + n] + VP[ii + 128 + n] + EP[ke + n];
                float g1 = gg[v] + VP[jj + 256 + n] + VP[ii + 384 + n] + EP[ke + 128 + n];
                float x  = c1 * scale_c + shift_c;
                float y  = g1 * scale_g + shift_g;
                float core = x / (1.0f + __expf(-x));      // silu
                float gate = 1.0f / (1.0f + __expf(-y));   // sigmoid
                atomicAdd(&S[(size_t)Lk[m] * 128 + n], core * gate);
            }
        }
        __syncthreads();
    }
}

// ---------------------------------------------------------------------------
// Kernel 6: d_out = S @ w_out^T + edge_feat
// ---------------------------------------------------------------------------
__global__ void out_gemm_kernel(const float* __restrict__ S,
                                const float* __restrict__ wout,
                                const float* __restrict__ ef,
                                float* __restrict__ out, int E) {
    __shared__ float As[16 * 130];
    const int m0   = blockIdx.x * 16;
    const int tid  = threadIdx.x;
    const int wave = tid >> 5, lane = tid & 31;

    for (int i = tid; i < 16 * BOND; i += 256) {
        int r = i >> 7, c = i & 127;
        int row = m0 + r; if (row >= E) row = E - 1;
        As[r * 130 + c] = S[(size_t)row * BOND + c];
    }
    __syncthreads();

    const int n0 = wave * 16;
    const int mlo = lane & 15, khalf = lane >> 4;
    const size_t wrow = (size_t)(n0 + mlo) * BOND;

    v8f acc = {};
    #pragma unroll
    for (int k = 0; k < BOND; k += 4) {
        const int kk = k + 2 * khalf;
        v2f a = ld2(&As[mlo * 130 + kk]);
        v2f b = ld2(wout + wrow + kk);
        acc = wmma4(a, b, acc);
    }
    #pragma unroll
    for (int v = 0; v < 8; ++v) {
        const int m = m0 + v + 8 * khalf;
        if (m < E) {
            const size_t o = (size_t)m * BOND + n0 + mlo;
            out[o] = acc[v] + ef[o];
        }
    }
}

// ---------------------------------------------------------------------------
// Host launcher
// ---------------------------------------------------------------------------
extern "C" void kernel_launch(void* const* d_in, const int* in_sizes, int n_in,
                              void* d_out, int out_size, void* d_ws, size_t ws_size,
                              hipStream_t stream) {
    const float* vf   = (const float*)d_in[0];
    const float* ef   = (const float*)d_in[1];
    const float* af   = (const float*)d_in[2];
    // d_in[3] = edge_index (unused by the reference math)
    const int* kidx   = (const int*)d_in[4];
    const int* jidx   = (const int*)d_in[5];
    const int* iidx   = (const int*)d_in[6];
    const float* w_cs = (const float*)d_in[7];
    const float* w_cd = (const float*)d_in[8];
    const float* w_cb = (const float*)d_in[9];
    const float* w_ca = (const float*)d_in[10];
    const float* w_gs = (const float*)d_in[11];
    const float* w_gd = (const float*)d_in[12];
    const float* w_gb = (const float*)d_in[13];
    const float* w_ga = (const float*)d_in[14];
    const float* bn_cg = (const float*)d_in[15];
    const float* bn_cb = (const float*)d_in[16];
    const float* bn_gg = (const float*)d_in[17];
    const float* bn_gb = (const float*)d_in[18];
    const float* w_out = (const float*)d_in[19];

    const int N = in_sizes[0] / ATOM;
    const int E = in_sizes[1] / BOND;
    const int T = in_sizes[4];

    float* ws    = (float*)d_ws;
    float* VP    = ws;                          // N * 512
    float* EP    = VP + (size_t)N * 512;        // E * 256
    float* S     = EP + (size_t)E * 256;        // E * 128
    float* stats = S  + (size_t)E * 128;        // 512
    float* bnp   = stats + 512;                 // 512

    hipMemsetAsync(S, 0, (size_t)E * 128 * sizeof(float), stream);
    hipMemsetAsync(stats, 0, 512 * sizeof(float), stream);

    const int nbN = (N + 15) / 16;
    const int nbE = (E + 15) / 16;

    node_proj_kernel<<<nbN, 256, 0, stream>>>(vf, w_cs, w_cd, w_gs, w_gd, VP, N);
    edge_proj_kernel<<<nbE, 256, 0, stream>>>(ef, w_cb, w_gb, EP, E);
    pass1_kernel<<<2048, 256, 0, stream>>>(af, kidx, jidx, iidx, w_ca, w_ga,
                                           VP, EP, stats, T);
    finalize_kernel<<<1, 128, 0, stream>>>(stats, bn_cg, bn_cb, bn_gg, bn_gb,
                                           bnp, 1.0f / (float)T);
    pass2_kernel<<<2048, 256, 0, stream>>>(af, kidx, jidx, iidx, w_ca, w_ga,
                                           VP, EP, bnp, S, T);
    out_gemm_kernel<<<nbE, 256, 0, stream>>>(S, w_out, ef, (float*)d_out, E);
}